// AttentionLayer_10471130268282
// MI455X (gfx1250) — compile-verified
//
#include <hip/hip_runtime.h>
#include <hip/hip_bf16.h>

typedef __attribute__((ext_vector_type(2))) float v2f;
typedef __attribute__((ext_vector_type(8))) float v8f;

#define B_  8
#define TD_ 128
#define TE_ 256
#define H_  512

#define WMMA_F32(a, b, c) \
    __builtin_amdgcn_wmma_f32_16x16x4_f32(false, (a), false, (b), (short)0, (c), false, false)

// ---------------------------------------------------------------------------
// GEMM: Y[M,512] = X[M,K] @ W[K,512], fp32, V_WMMA_F32_16X16X4_F32.
// blockDim = 256 (8 waves). Each wave computes a 16(M) x 64(N) strip with 4
// accumulators, reusing one A fragment across 4 WMMAs. 8 waves cover N=512.
// grid = (M/16).
// ---------------------------------------------------------------------------
__global__ __launch_bounds__(256)
void gemm_f32_wmma(const float* __restrict__ X, const float* __restrict__ W,
                   float* __restrict__ Y, int K) {
    const int N = H_;
    const int lane = threadIdx.x & 31;
    const int wave = threadIdx.x >> 5;
    const int m0 = blockIdx.x * 16;
    const int n0 = wave * 64;            // 8 waves * 64 = 512 = N
    const int m    = lane & 15;
    const int n    = lane & 15;
    const int koff = (lane >> 4) << 1;   // 0 (lanes 0-15) or 2 (lanes 16-31)

    v8f acc0 = {}, acc1 = {}, acc2 = {}, acc3 = {};
    const float* xrow = X + (size_t)(m0 + m) * K + koff;

    for (int kb = 0; kb < K; kb += 4) {
        v2f a;
        a.x = xrow[kb];
        a.y = xrow[kb + 1];

        const float* w0 = W + (size_t)(kb + koff) * N + n;   // row kb+koff
        const float* w1 = w0 + N;                            // row kb+koff+1
        __builtin_prefetch(w0 + 8 * N, 0, 1);                // global_prefetch_b8

        v2f b0, b1, b2, b3;
        b0.x = w0[n0];      b0.y = w1[n0];
        b1.x = w0[n0 + 16]; b1.y = w1[n0 + 16];
        b2.x = w0[n0 + 32]; b2.y = w1[n0 + 32];
        b3.x = w0[n0 + 48]; b3.y = w1[n0 + 48];

        acc0 = WMMA_F32(a, b0, acc0);
        acc1 = WMMA_F32(a, b1, acc1);
        acc2 = WMMA_F32(a, b2, acc2);
        acc3 = WMMA_F32(a, b3, acc3);
    }

    const int rowbase = m0 + ((lane >> 4) << 3);   // lanes 16-31 hold rows M+8
    #pragma unroll
    for (int r = 0; r < 8; ++r) {
        float* yrow = Y + (size_t)(rowbase + r) * N + n0 + n;
        yrow[0]  = acc0[r];
        yrow[16] = acc1[r];
        yrow[32] = acc2[r];
        yrow[48] = acc3[r];
    }
}

// ---------------------------------------------------------------------------
// score + masked softmax. One block per (b,t) row. 256 threads = 8 waves.
// Each wave computes one score s (lanes partition h in float4 chunks,
// shuffle-reduce), then the block does a 256-wide softmax (blockDim == TE_).
// tanh(x) = 1 - 2/(exp(2x)+1) via v_exp_f32 / v_rcp_f32 (branch-free,
// saturates correctly at +-1).
// ---------------------------------------------------------------------------
__device__ __forceinline__ float tanh_fast(float x) {
    float e = __builtin_amdgcn_exp2f(x * 2.8853900817779268f);  // e^{2x}
    return 1.0f - 2.0f * __builtin_amdgcn_rcpf(e + 1.0f);
}

__global__ __launch_bounds__(256)
void score_softmax_kernel(const float* __restrict__ ew,   // [B*TE, H]
                          const float* __restrict__ dw,   // [B*TD, H]
                          const int*   __restrict__ mask, // [B, TE]
                          const float* __restrict__ V,    // [H]
                          float*       __restrict__ attn) // [B*TD, TE]
{
    __shared__ float dws[H_];
    __shared__ float vs[H_];
    __shared__ float sc[TE_];
    __shared__ float red[256];

    const int tid  = threadIdx.x;
    const int lane = tid & 31;
    const int wave = tid >> 5;
    const int row  = blockIdx.x;          // b*TD + t
    const int b    = row / TD_;

    for (int i = tid; i < H_; i += 256) {
        dws[i] = dw[(size_t)row * H_ + i];
        vs[i]  = V[i];
    }
    __syncthreads();

    for (int s = wave; s < TE_; s += 8) {
        const float* er = ew + ((size_t)b * TE_ + s) * H_;
        float p = 0.0f;
        #pragma unroll
        for (int jj = 0; jj < H_ / 128; ++jj) {
            const int h = jj * 128 + lane * 4;
            float4 ev = *(const float4*)(er  + h);   // coalesced b128
            float4 dv = *(const float4*)(dws + h);   // ds_load_b128
            float4 vv = *(const float4*)(vs  + h);
            p += vv.x * tanh_fast(ev.x + dv.x);
            p += vv.y * tanh_fast(ev.y + dv.y);
            p += vv.z * tanh_fast(ev.z + dv.z);
            p += vv.w * tanh_fast(ev.w + dv.w);
        }
        #pragma unroll
        for (int off = 16; off > 0; off >>= 1)
            p += __shfl_xor(p, off, 32);
        if (lane == 0) {
            float msk = mask[b * TE_ + s] ? 0.0f : -1e9f;
            sc[s] = p + msk;
        }
    }
    __syncthreads();

    // softmax over TE_ == blockDim scores
    float v = sc[tid];
    red[tid] = v;
    __syncthreads();
    #pragma unroll
    for (int off = 128; off > 0; off >>= 1) {
        if (tid < off) red[tid] = fmaxf(red[tid], red[tid + off]);
        __syncthreads();
    }
    const float mx = red[0];
    __syncthreads();
    float e = __builtin_amdgcn_exp2f((v - mx) * 1.4426950408889634f);
    red[tid] = e;
    __syncthreads();
    #pragma unroll
    for (int off = 128; off > 0; off >>= 1) {
        if (tid < off) red[tid] += red[tid + off];
        __syncthreads();
    }
    const float inv = __builtin_amdgcn_rcpf(red[0]);
    attn[(size_t)row * TE_ + tid] = e * inv;
}

// ---------------------------------------------------------------------------
// context[b] = attn[b] @ enc[b]:  [TD,TE] @ [TE,H] per batch, WMMA f32.
// Same 16x64-per-wave register blocking; 8 waves cover H=512.
// grid = (TD/16, B).
// ---------------------------------------------------------------------------
__global__ __launch_bounds__(256)
void context_wmma(const float* __restrict__ attn,  // [B, TD, TE]
                  const float* __restrict__ enc,   // [B, TE, H]
                  float* __restrict__ ctx)         // [B, TD, H]
{
    const int N = H_;
    const int lane = threadIdx.x & 31;
    const int wave = threadIdx.x >> 5;
    const int b  = blockIdx.y;
    const int m0 = blockIdx.x * 16;      // over TD
    const int n0 = wave * 64;            // over H
    const int m    = lane & 15;
    const int n    = lane & 15;
    const int koff = (lane >> 4) << 1;

    const float* A = attn + (size_t)b * TD_ * TE_;
    const float* E = enc  + (size_t)b * TE_ * H_;

    v8f acc0 = {}, acc1 = {}, acc2 = {}, acc3 = {};
    const float* arow = A + (size_t)(m0 + m) * TE_ + koff;

    for (int kb = 0; kb < TE_; kb += 4) {
        v2f a;
        a.x = arow[kb];
        a.y = arow[kb + 1];

        const float* e0 = E + (size_t)(kb + koff) * N + n;
        const float* e1 = e0 + N;

        v2f b0, b1, b2, b3;
        b0.x = e0[n0];      b0.y = e1[n0];
        b1.x = e0[n0 + 16]; b1.y = e1[n0 + 16];
        b2.x = e0[n0 + 32]; b2.y = e1[n0 + 32];
        b3.x = e0[n0 + 48]; b3.y = e1[n0 + 48];

        acc0 = WMMA_F32(a, b0, acc0);
        acc1 = WMMA_F32(a, b1, acc1);
        acc2 = WMMA_F32(a, b2, acc2);
        acc3 = WMMA_F32(a, b3, acc3);
    }

    const int rowbase = m0 + ((lane >> 4) << 3);
    #pragma unroll
    for (int r = 0; r < 8; ++r) {
        float* crow = ctx + ((size_t)b * TD_ + rowbase + r) * N + n0 + n;
        crow[0]  = acc0[r];
        crow[16] = acc1[r];
        crow[32] = acc2[r];
        crow[48] = acc3[r];
    }
}

// ---------------------------------------------------------------------------
extern "C" void kernel_launch(void* const* d_in, const int* in_sizes, int n_in,
                              void* d_out, int out_size, void* d_ws, size_t ws_size,
                              hipStream_t stream) {
    const float* enc  = (const float*)d_in[0];  // [B, TE, H]
    const float* dec  = (const float*)d_in[1];  // [B, TD, H]
    const int*   mask = (const int*)  d_in[2];  // [B, TE]
    const float* W1   = (const float*)d_in[3];  // [H, H]
    const float* W2   = (const float*)d_in[4];  // [H, H]
    const float* V    = (const float*)d_in[5];  // [H, 1]

    float* ctx  = (float*)d_out;                         // [B, TD, H]
    float* attn = ctx + (size_t)B_ * TD_ * H_;           // [B, TD, TE]

    float* ew = (float*)d_ws;                            // [B*TE, H]  (4 MB)
    float* dw = ew + (size_t)B_ * TE_ * H_;              // [B*TD, H]  (2 MB)

    // ew = enc @ W1   (M = B*TE = 2048)
    gemm_f32_wmma<<<dim3((B_ * TE_) / 16), 256, 0, stream>>>(enc, W1, ew, H_);
    // dw = dec @ W2   (M = B*TD = 1024)
    gemm_f32_wmma<<<dim3((B_ * TD_) / 16), 256, 0, stream>>>(dec, W2, dw, H_);
    // scores + masked softmax -> attn
    score_softmax_kernel<<<B_ * TD_, 256, 0, stream>>>(ew, dw, mask, V, attn);
    // context = attn @ enc (batched)
    context_wmma<<<dim3(TD_ / 16, B_), 256, 0, stream>>>(attn, enc, ctx);
}